// Voxelizer_16939351015723
// MI455X (gfx1250) — compile-verified
//
#include <hip/hip_runtime.h>
#include <cstdint>
#include <cstddef>

// ---- problem constants (match reference) ----
#define NPTS   300000
#define BATCH  4
#define GXD    512
#define GYD    512
#define NC     (GXD*GYD)        // 262144 cells (z-dim == 1)
#define MV     30000            // MAX_VOXELS
#define MP     20               // MAX_POINTS
#define TOT    (BATCH*NPTS)
#define NB     ((NPTS+255)/256) // blocks per batch for scans = 1172
#define SENTC  NC               // sentinel cell id for invalid points
// d_out layout (floats): pillars | npoints | coors
#define NP_OFF ((size_t)BATCH*MV*MP*5)       // 12,000,000
#define CO_OFF (NP_OFF + (size_t)BATCH*MV)   // 12,120,000

typedef unsigned int v4u __attribute__((ext_vector_type(4)));
typedef int          v8i __attribute__((ext_vector_type(8)));
typedef int          v4i __attribute__((ext_vector_type(4)));

// -------------------- utility fills --------------------
__global__ void k_fill_f32(float* __restrict__ p, float v, long n) {
  long i = (long)blockIdx.x * blockDim.x + threadIdx.x;
  long s = (long)gridDim.x * blockDim.x;
  for (; i < n; i += s) p[i] = v;
}
__global__ void k_fill_i32(int* __restrict__ p, int v, long n) {
  long i = (long)blockIdx.x * blockDim.x + threadIdx.x;
  long s = (long)gridDim.x * blockDim.x;
  for (; i < n; i += s) p[i] = v;
}

// -------------------- cell id + counts --------------------
// Each block consumes a CONTIGUOUS 5120B tile of the points array -> stage it
// into LDS with ONE Tensor-Data-Mover descriptor (gfx1250 TDM), then compute
// voxel cells from LDS. tensor_dim0 clamps the tail block (OOB reads -> 0).
__global__ __launch_bounds__(256)
void k_lin(const float* __restrict__ pts, int* __restrict__ lin,
           int* __restrict__ cnt) {
  __shared__ float sh[256 * 5];
  const unsigned tileElems = 256u * 5u;                 // 1280 x 4B
  long blockElem = (long)blockIdx.x * (long)tileElems;

#if __has_builtin(__builtin_amdgcn_tensor_load_to_lds)
  if (threadIdx.x == 0) {
    unsigned lds = (unsigned)(size_t)(void*)sh;
    unsigned long long ga = (unsigned long long)(pts + blockElem);
    long remain = (long)TOT * 5 - blockElem;            // elements to end of input
    unsigned td0 = (remain > 0x7fffffffL) ? 0x7fffffffu : (unsigned)remain;

    v4u g0; v8i g1;
    v4i g2 = {0, 0, 0, 0}, g3 = {0, 0, 0, 0};
    v8i gx = {0, 0, 0, 0, 0, 0, 0, 0};
    // D# group0: count=1 | lds_addr | global_addr[56:0] | type=2 ("image")
    g0.x = 1u;
    g0.y = lds;
    g0.z = (unsigned)(ga & 0xffffffffu);
    g0.w = (unsigned)((ga >> 32) & 0x1ffffffu) | (2u << 30);
    // D# group1: wg_mask=0, data_size=4B, 1-row tile of tileElems elements,
    // tensor_dim0=td0 (tail OOB -> zero-fill), tensor_dim1=1.
    g1[0] = (int)(2u << 16);                                  // data_size=2 (4B)
    g1[1] = (int)((td0 & 0xffffu) << 16);                     // td0[15:0]
    g1[2] = (int)(((td0 >> 16) & 0xffffu) | (1u << 16));      // td0[31:16] | td1.lo=1
    g1[3] = (int)(tileElems << 16);                           // td1.hi=0 | tile_dim0
    g1[4] = 1;                                                // tile_dim1=1, tile_dim2=0
    g1[5] = (int)tileElems;                                   // dim0_stride (single row)
    g1[6] = 0;
    g1[7] = 0;
    __builtin_amdgcn_tensor_load_to_lds(g0, g1, g2, g3, gx, 0);
    __builtin_amdgcn_s_wait_tensorcnt(0);
  }
  __syncthreads();
#else
  // fallback: per-lane async gather of xyz into LDS
  {
    int gid0 = blockIdx.x * 256 + threadIdx.x;
    if (gid0 < TOT) {
      unsigned lds = (unsigned)(size_t)(void*)&sh[threadIdx.x * 5];
      unsigned long long g = (unsigned long long)(pts + (size_t)gid0 * 5);
      asm volatile(
          "global_load_async_to_lds_b32 %0, %1, off\n\t"
          "global_load_async_to_lds_b32 %0, %1, off offset:4\n\t"
          "global_load_async_to_lds_b32 %0, %1, off offset:8\n\t"
          "s_wait_asynccnt 0"
          :: "v"(lds), "v"(g) : "memory");
    }
    __syncthreads();
  }
#endif

  int gid = blockIdx.x * 256 + threadIdx.x;
  if (gid >= TOT) return;
  float x = sh[threadIdx.x * 5 + 0];
  float y = sh[threadIdx.x * 5 + 1];
  float z = sh[threadIdx.x * 5 + 2];

  int b = gid / NPTS;
  int i = gid - b * NPTS;
  if (i == 0) { x = 0.75f; y = 0.25f; z = 2.0f; }  // reference override

  float fx = floorf((x - (-51.2f)) / 0.2f);
  float fy = floorf((y - (-51.2f)) / 0.2f);
  float fz = floorf((z - (-5.0f)) / 8.0f);
  bool valid = (fx >= 0.f) && (fx < 512.f) &&
               (fy >= 0.f) && (fy < 512.f) &&
               (fz >= 0.f) && (fz < 1.f);
  if (valid) {
    int l = ((int)fz * GYD + (int)fy) * GXD + (int)fx;
    lin[gid] = l;
    atomicAdd(&cnt[b * NC + l], 1);     // order-independent
  } else {
    lin[gid] = SENTC;
  }
}

// -------------------- deterministic rank via atomicMax tournament ------------
// pass p in [1,20]: key = (p<<19) | (0x7FFFF - idx). Max key == smallest
// remaining point index in the cell for this pass. Stale keys from earlier
// passes always lose (smaller pass tag), so `best` needs no re-init.
__device__ __forceinline__ int rank_key(int pass, int i) {
  return (pass << 19) | (0x7FFFF - i);
}

__global__ __launch_bounds__(256)
void k_bid(const int* __restrict__ lin, const int* __restrict__ rank,
           int* __restrict__ best, int pass) {
  int gid = blockIdx.x * 256 + threadIdx.x;
  if (gid >= TOT) return;
  if (rank[gid] != -1) return;
  int l = lin[gid];
  if (l >= SENTC) return;
  int b = gid / NPTS;
  int i = gid - b * NPTS;
  atomicMax(&best[b * NC + l], rank_key(pass, i));
}

__global__ __launch_bounds__(256)
void k_claim(const int* __restrict__ lin, int* __restrict__ rank,
             const int* __restrict__ best, int pass) {
  int gid = blockIdx.x * 256 + threadIdx.x;
  if (gid >= TOT) return;
  if (rank[gid] != -1) return;
  int l = lin[gid];
  if (l >= SENTC) return;
  int b = gid / NPTS;
  int i = gid - b * NPTS;
  if (best[b * NC + l] == rank_key(pass, i)) rank[gid] = pass - 1;
}

// -------------------- two-level scan over is_first (rank==0) -----------------
__device__ __forceinline__ int block_incl_scan(int* sh, int v) {
  int tid = threadIdx.x;
  sh[tid] = v;
  __syncthreads();
  for (int off = 1; off < 256; off <<= 1) {
    int t = 0;
    if (tid >= off) t = sh[tid - off];
    __syncthreads();
    if (tid >= off) sh[tid] += t;
    __syncthreads();
  }
  return sh[tid];
}

__global__ __launch_bounds__(256)
void k_scan_blocks(const int* __restrict__ rank, int* __restrict__ partials) {
  __shared__ int sh[256];
  int b   = blockIdx.x / NB;
  int blk = blockIdx.x - b * NB;
  int i = blk * 256 + threadIdx.x;
  int f = 0;
  if (i < NPTS) f = (rank[b * NPTS + i] == 0) ? 1 : 0;
  sh[threadIdx.x] = f;
  __syncthreads();
  for (int s = 128; s > 0; s >>= 1) {
    if (threadIdx.x < s) sh[threadIdx.x] += sh[threadIdx.x + s];
    __syncthreads();
  }
  if (threadIdx.x == 0) partials[b * NB + blk] = sh[0];
}

__global__ __launch_bounds__(256)
void k_scan_partials(int* __restrict__ partials) {
  __shared__ int sh[256];
  int b = blockIdx.x;
  int run = 0;
  for (int base = 0; base < NB; base += 256) {
    int idx = base + threadIdx.x;
    int v = (idx < NB) ? partials[b * NB + idx] : 0;
    int incl = block_incl_scan(sh, v);
    int total = sh[255];
    if (idx < NB) partials[b * NB + idx] = run + incl - v;  // exclusive
    run += total;
    __syncthreads();
  }
}

__global__ __launch_bounds__(256)
void k_assign(const int* __restrict__ rank, const int* __restrict__ lin,
              const int* __restrict__ partials, int* __restrict__ slot_map,
              int* __restrict__ lin_of_slot) {
  __shared__ int sh[256];
  int b   = blockIdx.x / NB;
  int blk = blockIdx.x - b * NB;
  int i = blk * 256 + threadIdx.x;
  int f = 0;
  if (i < NPTS) f = (rank[b * NPTS + i] == 0) ? 1 : 0;
  int incl = block_incl_scan(sh, f);
  if (f) {
    int slot = partials[b * NB + blk] + incl - 1;  // first-occurrence order
    if (slot < MV) {
      int l = lin[b * NPTS + i];
      slot_map[b * NC + l] = slot;
      lin_of_slot[b * MV + slot] = l;
    }
  }
}

// -------------------- outputs --------------------
__global__ __launch_bounds__(256)
void k_coors_np(const int* __restrict__ lin_of_slot, const int* __restrict__ cnt,
                float* __restrict__ npo, float* __restrict__ coo) {
  int gid = blockIdx.x * 256 + threadIdx.x;
  if (gid >= BATCH * MV) return;
  int b = gid / MV;
  int l = lin_of_slot[gid];
  float np = 0.f, cz = -1.f, cy = -1.f, cx = -1.f;
  if (l >= 0) {
    int c = cnt[b * NC + l];
    np = (float)(c < MP ? c : MP);
    cz = (float)(l / (GXD * GYD));
    cy = (float)((l / GXD) % GYD);
    cx = (float)(l % GXD);
  }
  npo[gid] = np;
  coo[(size_t)gid * 4 + 0] = (float)b;
  coo[(size_t)gid * 4 + 1] = cz;
  coo[(size_t)gid * 4 + 2] = cy;
  coo[(size_t)gid * 4 + 3] = cx;
}

// Sparse gather of kept points: per-lane async global->LDS loads (each lane
// waits only on its own wave's ASYNCcnt; no barrier needed).
__global__ __launch_bounds__(256)
void k_scatter(const float* __restrict__ pts, const int* __restrict__ lin,
               const int* __restrict__ rank, const int* __restrict__ slot_map,
               float* __restrict__ pillars) {
  __shared__ float sh[256 * 5];
  int gid = blockIdx.x * 256 + threadIdx.x;
  if (gid >= TOT) return;
  int r = rank[gid];
  if (r < 0) return;                      // unranked => dropped (rank >= 20 or invalid)
  int b = gid / NPTS;
  int i = gid - b * NPTS;
  int s = slot_map[b * NC + lin[gid]];
  if (s >= MV) return;                    // overflow voxel => dropped

  unsigned lds = (unsigned)(size_t)(void*)&sh[threadIdx.x * 5];
  unsigned long long g = (unsigned long long)(pts + (size_t)gid * 5);
  asm volatile(
      "global_load_async_to_lds_b32 %0, %1, off\n\t"
      "global_load_async_to_lds_b32 %0, %1, off offset:4\n\t"
      "global_load_async_to_lds_b32 %0, %1, off offset:8\n\t"
      "global_load_async_to_lds_b32 %0, %1, off offset:12\n\t"
      "global_load_async_to_lds_b32 %0, %1, off offset:16\n\t"
      "s_wait_asynccnt 0"
      :: "v"(lds), "v"(g) : "memory");
  float f0 = sh[threadIdx.x * 5 + 0];
  float f1 = sh[threadIdx.x * 5 + 1];
  float f2 = sh[threadIdx.x * 5 + 2];
  float f3 = sh[threadIdx.x * 5 + 3];
  float f4 = sh[threadIdx.x * 5 + 4];
  if (i == 0) { f0 = 0.75f; f1 = 0.25f; f2 = 2.0f; }
  size_t o = (((size_t)b * MV + s) * MP + r) * 5;
  pillars[o + 0] = f0; pillars[o + 1] = f1; pillars[o + 2] = f2;
  pillars[o + 3] = f3; pillars[o + 4] = f4;
}

// -------------------- host orchestration --------------------
extern "C" void kernel_launch(void* const* d_in, const int* in_sizes, int n_in,
                              void* d_out, int out_size, void* d_ws, size_t ws_size,
                              hipStream_t stream) {
  const float* pts = (const float*)d_in[0];
  float* out = (float*)d_out;

  // workspace carve-up (int32 units), ~22.7 MB total
  int* ws = (int*)d_ws;
  size_t off = 0;
  int* lin         = ws + off; off += (size_t)TOT;
  int* rank        = ws + off; off += (size_t)TOT;
  int* cnt         = ws + off; off += (size_t)BATCH * NC;
  int* best        = ws + off; off += (size_t)BATCH * NC;
  int* slot_map    = ws + off; off += (size_t)BATCH * NC;
  int* lin_of_slot = ws + off; off += (size_t)BATCH * MV;
  int* partials    = ws + off; off += (size_t)BATCH * NB;
  (void)ws_size; (void)in_sizes; (void)n_in;

  const int B = 256;
  const int gTOT = (TOT + B - 1) / B;

  // init
  k_fill_f32<<<4096, B, 0, stream>>>(out, 0.f, (long)out_size);
  k_fill_i32<<<2048, B, 0, stream>>>(cnt, 0, (long)BATCH * NC);
  k_fill_i32<<<2048, B, 0, stream>>>(best, 0, (long)BATCH * NC);
  k_fill_i32<<<2048, B, 0, stream>>>(slot_map, MV, (long)BATCH * NC);
  k_fill_i32<<<512,  B, 0, stream>>>(lin_of_slot, -1, (long)BATCH * MV);
  k_fill_i32<<<2048, B, 0, stream>>>(rank, -1, (long)TOT);

  // cell ids + per-cell counts (TDM tile-staged into LDS)
  k_lin<<<gTOT, B, 0, stream>>>(pts, lin, cnt);

  // deterministic per-cell rank: 20 min-index tournament rounds
  for (int p = 1; p <= MP; ++p) {
    k_bid  <<<gTOT, B, 0, stream>>>(lin, rank, best, p);
    k_claim<<<gTOT, B, 0, stream>>>(lin, rank, best, p);
  }

  // first-occurrence slot numbering (prefix sum over rank==0 flags)
  k_scan_blocks  <<<BATCH * NB, B, 0, stream>>>(rank, partials);
  k_scan_partials<<<BATCH,      B, 0, stream>>>(partials);
  k_assign       <<<BATCH * NB, B, 0, stream>>>(rank, lin, partials, slot_map, lin_of_slot);

  // outputs
  k_coors_np<<<(BATCH * MV + B - 1) / B, B, 0, stream>>>(lin_of_slot, cnt,
                                                         out + NP_OFF, out + CO_OFF);
  k_scatter<<<gTOT, B, 0, stream>>>(pts, lin, rank, slot_map, out);
}